// GIN_3layer_11510512353341
// MI455X (gfx1250) — compile-verified
//
#include <hip/hip_runtime.h>
#include <hip/hip_bf16.h>

typedef __attribute__((ext_vector_type(2))) float v2f;
typedef __attribute__((ext_vector_type(8))) float v8f;

#define N_NODES 50000
#define N_EDGES 800000
#define CH 64
#define N_GRAPHS 128
#define OUT_CH 10
#define SA_PITCH 68   // 64 + 4 pad: keeps 16B alignment, spreads LDS banks

// ---------------------------------------------------------------------------
// Zero-fill helper (graph-capture safe; no hipMemsetAsync dependence)
// ---------------------------------------------------------------------------
__global__ void gin_zero(float* __restrict__ p, int n) {
    int i = blockIdx.x * blockDim.x + threadIdx.x;
    if (i < n) p[i] = 0.0f;
}

// ---------------------------------------------------------------------------
// Edge scatter-add: agg[dst] += H[src]. 16 threads per edge, 4 channels each.
// float4 gather (coalesced within the 16-thread group), 4x global atomic f32.
// ---------------------------------------------------------------------------
__global__ void gin_scatter(const float* __restrict__ H,
                            const int* __restrict__ srcs,
                            const int* __restrict__ dsts,
                            float* __restrict__ AGG) {
    long long t = (long long)blockIdx.x * blockDim.x + threadIdx.x;
    int e = (int)(t >> 4);
    if (e >= N_EDGES) return;
    int c4 = ((int)t & 15) * 4;
    int s = srcs[e];
    int d = dsts[e];
    const float4 v = *(const float4*)(H + (size_t)s * CH + c4);
    float* p = AGG + (size_t)d * CH + c4;
    atomicAdd(p + 0, v.x);
    atomicAdd(p + 1, v.y);
    atomicAdd(p + 2, v.z);
    atomicAdd(p + 3, v.w);
}

// ---------------------------------------------------------------------------
// Fused (X + AGG) @ W^T + b, ReLU  — f32 WMMA 16x16x4.
// Block = 128 threads (4 waves). Block computes rows [16*bid, 16*bid+16),
// wave w computes output columns [16w, 16w+16). K-loop: 16 WMMA ops.
// ---------------------------------------------------------------------------
__global__ __launch_bounds__(128)
void gin_gemm_relu(const float* __restrict__ X,
                   const float* __restrict__ AGG,
                   const float* __restrict__ W,      // [CH][CH] row-major: W[n][k]
                   const float* __restrict__ bias,   // [CH]
                   float* __restrict__ OUT) {
    __shared__ float sA[16][SA_PITCH];   // z = x + agg tile
    __shared__ float sW[CH][SA_PITCH];   // sW[n][k] = W[n][k] = B[k][n]

    const int tid  = threadIdx.x;
    const int wave = tid >> 5;
    const int lane = tid & 31;
    const int row0 = blockIdx.x * 16;

    // Stage W (4096 floats) into LDS, coalesced float4 per row.
    for (int i = tid; i < CH * (CH / 4); i += 128) {
        int n  = i >> 4;      // row 0..63
        int c4 = i & 15;      // float4 column 0..15
        float4 wv = ((const float4*)(W + (size_t)n * CH))[c4];
        sW[n][c4 * 4 + 0] = wv.x;
        sW[n][c4 * 4 + 1] = wv.y;
        sW[n][c4 * 4 + 2] = wv.z;
        sW[n][c4 * 4 + 3] = wv.w;
    }
    // Stage z = X + AGG tile (16 x 64), coalesced float4.
    for (int i = tid; i < 16 * (CH / 4); i += 128) {
        int r  = i >> 4;
        int c4 = i & 15;
        int gr = row0 + r;
        float4 xv = ((const float4*)(X   + (size_t)gr * CH))[c4];
        float4 av = ((const float4*)(AGG + (size_t)gr * CH))[c4];
        sA[r][c4 * 4 + 0] = xv.x + av.x;
        sA[r][c4 * 4 + 1] = xv.y + av.y;
        sA[r][c4 * 4 + 2] = xv.z + av.z;
        sA[r][c4 * 4 + 3] = xv.w + av.w;
    }
    __syncthreads();

    const int n0    = wave * 16;
    const int mlo   = lane & 15;           // row (A) / col (B,D) within tile
    const int khalf = (lane >> 4) * 2;     // lanes 16-31 carry K+2,K+3

    v8f acc = {};
    #pragma unroll
    for (int kb = 0; kb < CH; kb += 4) {
        v2f a, b;
        a.x = sA[mlo][kb + khalf];
        a.y = sA[mlo][kb + khalf + 1];
        b.x = sW[n0 + mlo][kb + khalf];
        b.y = sW[n0 + mlo][kb + khalf + 1];
        acc = __builtin_amdgcn_wmma_f32_16x16x4_f32(
            /*neg_a=*/false, a, /*neg_b=*/false, b,
            /*c_mod=*/(short)0, acc, /*reuse_a=*/false, /*reuse_b=*/false);
    }

    // D layout: VGPR v -> row (v + 8*(lane>=16)), col = lane&15.
    const int n  = n0 + mlo;
    const float bn = bias[n];
    const int mbase = row0 + ((lane >> 4) * 8);
    #pragma unroll
    for (int v = 0; v < 8; ++v) {
        float val = acc[v] + bn;
        OUT[(size_t)(mbase + v) * CH + n] = val > 0.0f ? val : 0.0f;
    }
}

// ---------------------------------------------------------------------------
// Pooling: sums[batch[node]] += H[node]; cnts[batch[node]] += 1.
// 16 threads per node, 4 channels each.
// ---------------------------------------------------------------------------
__global__ void gin_pool(const float* __restrict__ H,
                         const int* __restrict__ batch,
                         float* __restrict__ sums,
                         float* __restrict__ cnts) {
    long long t = (long long)blockIdx.x * blockDim.x + threadIdx.x;
    int node = (int)(t >> 4);
    if (node >= N_NODES) return;
    int c4 = ((int)t & 15) * 4;
    int g = batch[node];
    const float4 v = *(const float4*)(H + (size_t)node * CH + c4);
    float* p = sums + (size_t)g * CH + c4;
    atomicAdd(p + 0, v.x);
    atomicAdd(p + 1, v.y);
    atomicAdd(p + 2, v.z);
    atomicAdd(p + 3, v.w);
    if (((int)t & 15) == 0) atomicAdd(&cnts[g], 1.0f);
}

// ---------------------------------------------------------------------------
// Final: out[g][o] = (sums[g]/max(cnt,1)) . Wl[o] + bl[o]
// ---------------------------------------------------------------------------
__global__ void gin_final(const float* __restrict__ sums,
                          const float* __restrict__ cnts,
                          const float* __restrict__ Wl,   // [OUT_CH][CH]
                          const float* __restrict__ bl,
                          float* __restrict__ out) {
    int t = blockIdx.x * blockDim.x + threadIdx.x;
    if (t >= N_GRAPHS * OUT_CH) return;
    int g = t / OUT_CH;
    int o = t % OUT_CH;
    float c = cnts[g];
    float inv = 1.0f / (c > 1.0f ? c : 1.0f);
    float acc = 0.0f;
    for (int k = 0; k < CH; ++k)
        acc += (sums[g * CH + k] * inv) * Wl[o * CH + k];
    out[t] = acc + bl[o];
}

// ---------------------------------------------------------------------------
extern "C" void kernel_launch(void* const* d_in, const int* in_sizes, int n_in,
                              void* d_out, int out_size, void* d_ws, size_t ws_size,
                              hipStream_t stream) {
    const float* x  = (const float*)d_in[0];
    const float* W1 = (const float*)d_in[1];
    const float* b1 = (const float*)d_in[2];
    const float* W2 = (const float*)d_in[3];
    const float* b2 = (const float*)d_in[4];
    const float* W3 = (const float*)d_in[5];
    const float* b3 = (const float*)d_in[6];
    const float* Wl = (const float*)d_in[7];
    const float* bl = (const float*)d_in[8];
    const int*   ei = (const int*)d_in[9];     // [2, N_EDGES] (int32 under JAX default)
    const int*   bt = (const int*)d_in[10];    // [N_NODES]
    float* out = (float*)d_out;

    const int* srcs = ei;
    const int* dsts = ei + N_EDGES;

    // Workspace layout (floats)
    float* ws   = (float*)d_ws;
    float* agg  = ws;                                  // N_NODES*CH
    float* hA   = agg + (size_t)N_NODES * CH;          // N_NODES*CH
    float* hB   = hA  + (size_t)N_NODES * CH;          // N_NODES*CH
    float* sums = hB  + (size_t)N_NODES * CH;          // N_GRAPHS*CH
    float* cnts = sums + (size_t)N_GRAPHS * CH;        // N_GRAPHS

    const int nAgg      = N_NODES * CH;
    const int zeroBlk   = 256;
    const int zeroGrid  = (nAgg + zeroBlk - 1) / zeroBlk;
    const int scatGrid  = (N_EDGES * 16 + 255) / 256;
    const int gemmGrid  = N_NODES / 16;                // 3125, exact
    const int poolGrid  = (N_NODES * 16 + 255) / 256;

    // Layer 1
    gin_zero<<<zeroGrid, zeroBlk, 0, stream>>>(agg, nAgg);
    gin_scatter<<<scatGrid, 256, 0, stream>>>(x, srcs, dsts, agg);
    gin_gemm_relu<<<gemmGrid, 128, 0, stream>>>(x, agg, W1, b1, hA);

    // Layer 2
    gin_zero<<<zeroGrid, zeroBlk, 0, stream>>>(agg, nAgg);
    gin_scatter<<<scatGrid, 256, 0, stream>>>(hA, srcs, dsts, agg);
    gin_gemm_relu<<<gemmGrid, 128, 0, stream>>>(hA, agg, W2, b2, hB);

    // Layer 3
    gin_zero<<<zeroGrid, zeroBlk, 0, stream>>>(agg, nAgg);
    gin_scatter<<<scatGrid, 256, 0, stream>>>(hB, srcs, dsts, agg);
    gin_gemm_relu<<<gemmGrid, 128, 0, stream>>>(hB, agg, W3, b3, hA);

    // Pool + final linear
    gin_zero<<<(N_GRAPHS * CH + N_GRAPHS + 255) / 256, 256, 0, stream>>>(sums, N_GRAPHS * CH + N_GRAPHS);
    gin_pool<<<poolGrid, 256, 0, stream>>>(hA, bt, sums, cnts);
    gin_final<<<(N_GRAPHS * OUT_CH + 127) / 128, 128, 0, stream>>>(sums, cnts, Wl, bl, out);
}